// VQLocalProbAvgPool_71829033058531
// MI455X (gfx1250) — compile-verified
//
#include <hip/hip_runtime.h>

// Problem constants (from reference): B=32, N=2, L=2048, D=1024, V=320
#define B_      32
#define NN      2
#define LL      2048
#define DD      1024
#define VV      320
#define TPB     256
#define SPLITS  32
#define RPB     (LL / SPLITS)   // 64 rows per block
#define STG     4               // rows per async stage
#define NSTAGE  (RPB / STG)     // 16 stages

// ---------------------------------------------------------------------------
// Kernel 1: per-batch VQ histogram + normalized inverse-frequency weights.
// One block per batch. LDS atomics (ds_add) for the two 320-bin histograms,
// wave32 shuffle reduction for the normalizer.
// ---------------------------------------------------------------------------
__global__ __launch_bounds__(TPB) void vq_weights_kernel(
    const int* __restrict__ lens, const int* __restrict__ vq,
    float* __restrict__ w)
{
  const int b   = blockIdx.x;
  const int tid = threadIdx.x;

  __shared__ unsigned cx[VV];
  __shared__ unsigned cy[VV];
  __shared__ float    probs[LL];
  __shared__ float    partsum[TPB / 32];

  for (int i = tid; i < VV; i += TPB) { cx[i] = 0u; cy[i] = 0u; }
  __syncthreads();

  const int* vqb = vq + (size_t)b * LL * 2;
  for (int l = tid; l < LL; l += TPB) {
    atomicAdd(&cx[vqb[2 * l + 0]], 1u);   // ds_add_u32
    atomicAdd(&cy[vqb[2 * l + 1]], 1u);
  }
  __syncthreads();

  const int len = lens[b];
  float local = 0.0f;
  for (int l = tid; l < LL; l += TPB) {
    const unsigned f = cx[vqb[2 * l + 0]] + cy[vqb[2 * l + 1]];
    const float p = (l < len) ? (1.0f / (float)f) : 0.0f;
    probs[l] = p;
    local += p;
  }

  // wave32 tree reduction, then cross-wave combine
  #pragma unroll
  for (int off = 16; off > 0; off >>= 1)
    local += __shfl_down(local, off, 32);
  if ((tid & 31) == 0) partsum[tid >> 5] = local;
  __syncthreads();

  float tot = 0.0f;
  #pragma unroll
  for (int i = 0; i < TPB / 32; ++i) tot += partsum[i];
  const float inv = 1.0f / tot;

  for (int l = tid; l < LL; l += TPB)
    w[(size_t)b * LL + l] = probs[l] * inv;
}

// ---------------------------------------------------------------------------
// Kernel 2: bandwidth-bound weighted row reduction over an L-split.
// feat rows are streamed HBM -> LDS with gfx1250 async copies
// (global_load_async_to_lds_b128), double buffered via s_wait_asynccnt.
// Each thread owns one float4 of D; partial results per split (deterministic).
// ---------------------------------------------------------------------------
__global__ __launch_bounds__(TPB) void vq_gemv_partial_kernel(
    const float* __restrict__ feat_all, const float* __restrict__ w,
    float* __restrict__ part)
{
  const int tid = threadIdx.x;
  const int b   = blockIdx.y;
  const int r0  = blockIdx.x * RPB;

  // take the n = N-1 slice of input_feature (B, N, L, D)
  const float* feat = feat_all + ((size_t)b * NN + (NN - 1)) * (size_t)LL * DD;
  const float* wrow = w + (size_t)b * LL + r0;
  const int col = tid * 4;   // 256 threads * float4 == 1024 = D

  __shared__ float4 tile[2][STG][TPB];   // 32 KB, double buffered

  float4 acc = make_float4(0.f, 0.f, 0.f, 0.f);

  auto issue_stage = [&](int buf, int stage) {
    #pragma unroll
    for (int j = 0; j < STG; ++j) {
      // low 32 bits of the generic pointer to a __shared__ object == LDS byte
      // offset, which is what the instruction's VDST operand expects
      unsigned ldsoff = (unsigned)(size_t)&tile[buf][j][tid];
      const float* g = feat + (size_t)(r0 + stage * STG + j) * DD + col;
      asm volatile("global_load_async_to_lds_b128 %0, %1, off"
                   :: "v"(ldsoff), "v"(g) : "memory");
    }
  };

  issue_stage(0, 0);                       // prologue: ASYNCcnt = 4
  for (int s = 0; s < NSTAGE; ++s) {
    const int buf = s & 1;
    if (s + 1 < NSTAGE) {
      issue_stage(buf ^ 1, s + 1);         // ASYNCcnt = 8
      asm volatile("s_wait_asynccnt 4" ::: "memory");  // stage s landed in LDS
    } else {
      asm volatile("s_wait_asynccnt 0" ::: "memory");
    }
    #pragma unroll
    for (int j = 0; j < STG; ++j) {
      const float  wl = wrow[s * STG + j]; // uniform -> scalar load
      const float4 v  = tile[buf][j][tid]; // ds_load_b128
      acc.x = fmaf(wl, v.x, acc.x);
      acc.y = fmaf(wl, v.y, acc.y);
      acc.z = fmaf(wl, v.z, acc.z);
      acc.w = fmaf(wl, v.w, acc.w);
    }
  }

  float4* pout = (float4*)(part + ((size_t)blockIdx.x * B_ + b) * DD + col);
  *pout = acc;
}

// ---------------------------------------------------------------------------
// Kernel 3: ordered (deterministic) sum of the SPLITS partials -> d_out.
// ---------------------------------------------------------------------------
__global__ __launch_bounds__(TPB) void vq_reduce_kernel(
    const float* __restrict__ part, float* __restrict__ out)
{
  const int i = blockIdx.x * TPB + threadIdx.x;
  if (i >= B_ * DD) return;
  float s = 0.0f;
  #pragma unroll 4
  for (int k = 0; k < SPLITS; ++k)
    s += part[(size_t)k * (B_ * DD) + i];
  out[i] = s;
}

// ---------------------------------------------------------------------------
extern "C" void kernel_launch(void* const* d_in, const int* in_sizes, int n_in,
                              void* d_out, int out_size, void* d_ws, size_t ws_size,
                              hipStream_t stream)
{
  (void)in_sizes; (void)n_in; (void)out_size; (void)ws_size;

  const float* feat = (const float*)d_in[0];   // (B, N, L, D) f32
  const int*   lens = (const int*)d_in[1];     // (B,) int
  const int*   vq   = (const int*)d_in[2];     // (B, L, 2) int

  // workspace layout: [ weights B*L f32 | partials SPLITS*B*D f32 ]
  float* w    = (float*)d_ws;
  float* part = w + (size_t)B_ * LL;

  vq_weights_kernel<<<B_, TPB, 0, stream>>>(lens, vq, w);
  vq_gemv_partial_kernel<<<dim3(SPLITS, B_), TPB, 0, stream>>>(feat, w, part);
  vq_reduce_kernel<<<(B_ * DD + TPB - 1) / TPB, TPB, 0, stream>>>(part, (float*)d_out);
}